// Relational_GCN_56899726737496
// MI455X (gfx1250) — compile-verified
//
#include <hip/hip_runtime.h>
#include <hip/hip_bf16.h>

// ---------------------------------------------------------------------------
// Relational GCN for MI455X (gfx1250, wave32, WMMA).
// Per-relation GEMM (keeps the 25.6MB relation buffer L2-resident) + fused
// edge scatter-add. GEMM uses v_wmma_f32_16x16x4_f32 with the 128x128 weight
// staged TRANSPOSED in LDS (pitch 132 -> conflict-free ds_load_b64 B-frags).
// ---------------------------------------------------------------------------

#define NN   50000
#define EE   800000
#define INF_ 128
#define HH   128
#define RR   16
#define BB   8
#define CC   10
#define NEG_SLOPE 0.01f
#define WPITCH 132  // transposed-W LDS pitch: banks (4*col+kk)%64 all distinct

typedef float v2f __attribute__((ext_vector_type(2)));
typedef float v8f __attribute__((ext_vector_type(8)));

// ---------------------------------------------------------------------------
// W[r,i,o] = sum_b coeff[r,b] * basis[b,i,o]      (IO = IN*H flattened)
// ---------------------------------------------------------------------------
__global__ void build_w_kernel(const float* __restrict__ coeff,
                               const float* __restrict__ basis,
                               float* __restrict__ W, int IO) {
    int idx = blockIdx.x * blockDim.x + threadIdx.x;
    int total = RR * IO;
    if (idx >= total) return;
    int r = idx / IO;
    int io = idx - r * IO;
    float s = 0.f;
#pragma unroll
    for (int b = 0; b < BB; ++b)
        s += coeff[r * BB + b] * basis[b * IO + io];
    W[idx] = s;
}

__global__ void zero_kernel(float* __restrict__ p, int n) {
    int stride = gridDim.x * blockDim.x;
    for (int i = blockIdx.x * blockDim.x + threadIdx.x; i < n; i += stride)
        p[i] = 0.f;
}

// ---------------------------------------------------------------------------
// GEMM: C[M x 128] = A[M x 128] @ W[128 x 128] with epilogue.
//   MODE 0: C = A@W
//   MODE 1: C = leaky(agg + A@W + bias)
//   MODE 2: C = agg + A@W + bias
// 256 threads = 8 waves; each wave computes a 16-row x 128-col strip using
// v_wmma_f32_16x16x4_f32. W staged transposed in LDS:
//   sWt[o * WPITCH + i] = W[i][o]
// so a B fragment (rows kk,kk+1 of column c) is one contiguous 8B ds_load_b64.
// ---------------------------------------------------------------------------
template <int MODE>
__global__ void gemm128_kernel(const float* __restrict__ A,
                               const float* __restrict__ W,
                               const float* __restrict__ agg,
                               const float* __restrict__ bias,
                               float* __restrict__ C, int M) {
    __shared__ float sWt[128 * WPITCH];  // 67.6 KB

    // cooperative transposed staging: read W row-major (coalesced float4),
    // write columns. One-time cost per block.
    {
        const float4* Wv = (const float4*)W;
        for (int i = threadIdx.x; i < 128 * 32; i += blockDim.x) {
            int row = i >> 5;             // K index 0..127
            int c4 = (i & 31) << 2;       // column group
            float4 w = Wv[i];             // W[row][c4..c4+3]
            sWt[(c4 + 0) * WPITCH + row] = w.x;
            sWt[(c4 + 1) * WPITCH + row] = w.y;
            sWt[(c4 + 2) * WPITCH + row] = w.z;
            sWt[(c4 + 3) * WPITCH + row] = w.w;
        }
    }
    __syncthreads();

    const int wave = threadIdx.x >> 5;
    const int lane = threadIdx.x & 31;
    const int m0 = blockIdx.x * 128 + wave * 16;
    if (m0 >= M) return;  // wave-uniform exit: EXEC stays all-1s for live waves

    // A fragment addressing (16x4 f32 layout):
    //   lanes 0-15  -> rows M=0..15, K = k+0 / k+1
    //   lanes 16-31 -> rows M=0..15, K = k+2 / k+3
    int arow = m0 + (lane & 15);
    if (arow >= M) arow = M - 1;  // clamp: keeps EXEC full; stores are guarded
    const int koff = 2 * (lane >> 4);
    const int col = lane & 15;
    const float* Aptr = A + (size_t)arow * 128 + koff;  // 8B-aligned pairs

    v8f acc[8];
#pragma unroll
    for (int t = 0; t < 8; ++t) acc[t] = {};

    // software-pipelined A: prefetch next float2 before the WMMA burst
    v2f a_nxt;
    a_nxt.x = Aptr[0];
    a_nxt.y = Aptr[1];
#pragma unroll 4
    for (int k = 0; k < 128; k += 4) {
        const v2f a = a_nxt;
        if (k + 4 < 128) {
            a_nxt.x = Aptr[k + 4];
            a_nxt.y = Aptr[k + 5];
        }
        const int kk = k + koff;  // even -> 8B-aligned ds_load_b64
#pragma unroll
        for (int t = 0; t < 8; ++t) {
            const v2f b = *(const v2f*)&sWt[(t * 16 + col) * WPITCH + kk];
            acc[t] = __builtin_amdgcn_wmma_f32_16x16x4_f32(
                false, a, false, b, (short)0, acc[t], false, false);
        }
    }

    // C layout (16x16 f32): VGPR j holds row m0+j (lanes 0-15) or m0+8+j
    // (lanes 16-31), column = (lane&15) within the 16-col tile.
    const int rbase = m0 + ((lane >> 4) ? 8 : 0);
#pragma unroll
    for (int t = 0; t < 8; ++t) {
        const int c = t * 16 + col;
#pragma unroll
        for (int j = 0; j < 8; ++j) {
            const int row = rbase + j;
            if (row >= M) continue;
            float v = acc[t][j];
            if (MODE != 0) {
                v += agg[(size_t)row * 128 + c] + bias[c];
                if (MODE == 1) v = (v > 0.f) ? v : v * NEG_SLOPE;
            }
            C[(size_t)row * 128 + c] = v;
        }
    }
}

// ---------------------------------------------------------------------------
// Edge scatter for one relation: agg[dst[e]] += buf[src[e]] for etype[e]==r.
// One wave per edge (edge index is wave-uniform -> scalar loads of indices).
// Both buf (25.6MB) and agg (25.6MB) are L2-resident on MI455X (192MB L2).
// ---------------------------------------------------------------------------
__global__ void edge_scatter_kernel(const float* __restrict__ buf,
                                    const int* __restrict__ src,
                                    const int* __restrict__ dst,
                                    const int* __restrict__ etype,
                                    int r, float* __restrict__ agg) {
    const int wpb = blockDim.x >> 5;
    const int w = blockIdx.x * wpb + (threadIdx.x >> 5);
    const int lane = threadIdx.x & 31;
    const int stride = gridDim.x * wpb;
    for (int e = w; e < EE; e += stride) {
        if (etype[e] != r) continue;
        const float* s = buf + (size_t)src[e] * 128;
        float* d = agg + (size_t)dst[e] * 128;
#pragma unroll
        for (int j = 0; j < 4; ++j)
            atomicAdd(d + lane + j * 32, s[lane + j * 32]);
    }
}

// ---------------------------------------------------------------------------
// hv[n] = dot(h2[n,:], wagg) + bagg
// ---------------------------------------------------------------------------
__global__ void node_dot_kernel(const float* __restrict__ h,
                                const float* __restrict__ wagg,
                                const float* __restrict__ bagg,
                                float* __restrict__ out) {
    int n = blockIdx.x * blockDim.x + threadIdx.x;
    if (n >= NN) return;
    const float4* row = (const float4*)(h + (size_t)n * 128);
    const float4* wv = (const float4*)wagg;
    float acc = 0.f;
#pragma unroll
    for (int j = 0; j < 32; ++j) {
        float4 a = row[j], b = wv[j];
        acc += a.x * b.x + a.y * b.y + a.z * b.z + a.w * b.w;
    }
    out[n] = acc + bagg[0];
}

// ---------------------------------------------------------------------------
// d1[j] = sum_n hv[n] * wd1[n,j] + bd1[j]        (j = blockIdx.x, 100 blocks)
// ---------------------------------------------------------------------------
__global__ void dense1_kernel(const float* __restrict__ hv,
                              const float* __restrict__ wd1,
                              const float* __restrict__ bd1,
                              float* __restrict__ d1) {
    __shared__ float red[256];
    const int j = blockIdx.x;
    float acc = 0.f;
    for (int n = threadIdx.x; n < NN; n += blockDim.x)
        acc += hv[n] * wd1[(size_t)n * 100 + j];
    red[threadIdx.x] = acc;
    __syncthreads();
    for (int s = 128; s > 0; s >>= 1) {
        if ((int)threadIdx.x < s) red[threadIdx.x] += red[threadIdx.x + s];
        __syncthreads();
    }
    if (threadIdx.x == 0) d1[j] = red[0] + bd1[j];
}

// ---------------------------------------------------------------------------
// d2 = leaky(d1 @ wd2 + bd2)  [20];   out = d2 @ wd3 + bd3  [10]
// ---------------------------------------------------------------------------
__global__ void head_kernel(const float* __restrict__ d1,
                            const float* __restrict__ wd2,
                            const float* __restrict__ bd2,
                            const float* __restrict__ wd3,
                            const float* __restrict__ bd3,
                            float* __restrict__ out) {
    __shared__ float d2[20];
    const int t = threadIdx.x;
    if (t < 20) {
        float acc = 0.f;
        for (int k = 0; k < 100; ++k) acc += d1[k] * wd2[k * 20 + t];
        acc += bd2[t];
        d2[t] = (acc > 0.f) ? acc : acc * NEG_SLOPE;
    }
    __syncthreads();
    if (t < 10) {
        float acc = 0.f;
        for (int k = 0; k < 20; ++k) acc += d2[k] * wd3[k * 10 + t];
        out[t] = acc + bd3[t];
    }
}

// ---------------------------------------------------------------------------
extern "C" void kernel_launch(void* const* d_in, const int* in_sizes, int n_in,
                              void* d_out, int out_size, void* d_ws,
                              size_t ws_size, hipStream_t stream) {
    const float* x      = (const float*)d_in[0];
    const int*   src    = (const int*)d_in[1];
    const int*   dst    = (const int*)d_in[2];
    const int*   etype  = (const int*)d_in[3];
    const float* coeff1 = (const float*)d_in[4];
    const float* basis1 = (const float*)d_in[5];
    const float* wloop1 = (const float*)d_in[6];
    const float* b1     = (const float*)d_in[7];
    const float* coeff2 = (const float*)d_in[8];
    const float* basis2 = (const float*)d_in[9];
    const float* wloop2 = (const float*)d_in[10];
    const float* b2     = (const float*)d_in[11];
    const float* wagg   = (const float*)d_in[12];
    const float* bagg   = (const float*)d_in[13];
    const float* wd1    = (const float*)d_in[14];
    const float* bd1    = (const float*)d_in[15];
    const float* wd2    = (const float*)d_in[16];
    const float* bd2    = (const float*)d_in[17];
    const float* wd3    = (const float*)d_in[18];
    const float* bd3    = (const float*)d_in[19];
    float* out = (float*)d_out;

    // workspace layout (floats): ~131 MB total
    float* ws = (float*)d_ws;
    size_t off = 0;
    float* W1   = ws + off; off += (size_t)RR * INF_ * HH;  // 1 MB
    float* W2   = ws + off; off += (size_t)RR * HH * HH;    // 1 MB
    float* buf  = ws + off; off += (size_t)NN * HH;         // 25.6 MB
    float* agg1 = ws + off; off += (size_t)NN * HH;
    float* agg2 = ws + off; off += (size_t)NN * HH;
    float* h1   = ws + off; off += (size_t)NN * HH;
    float* h2   = ws + off; off += (size_t)NN * HH;
    float* hv   = ws + off; off += (size_t)NN;
    float* d1v  = ws + off; off += 128;

    const int NH = NN * HH;
    const dim3 blk256(256);
    const dim3 gemmGrid((NN + 127) / 128);

    // build relation weight matrices
    build_w_kernel<<<dim3((RR * INF_ * HH + 255) / 256), blk256, 0, stream>>>(
        coeff1, basis1, W1, INF_ * HH);
    build_w_kernel<<<dim3((RR * HH * HH + 255) / 256), blk256, 0, stream>>>(
        coeff2, basis2, W2, HH * HH);

    // zero accumulators
    zero_kernel<<<dim3(2048), blk256, 0, stream>>>(agg1, NH);
    zero_kernel<<<dim3(2048), blk256, 0, stream>>>(agg2, NH);

    // ---- layer 1: per-relation GEMM (L2-resident buf) + edge scatter ----
    for (int r = 0; r < RR; ++r) {
        gemm128_kernel<0><<<gemmGrid, blk256, 0, stream>>>(
            x, W1 + (size_t)r * INF_ * HH, nullptr, nullptr, buf, NN);
        edge_scatter_kernel<<<dim3(2048), blk256, 0, stream>>>(
            buf, src, dst, etype, r, agg1);
    }
    // h1 = leaky(agg1 + x @ wloop1 + b1)
    gemm128_kernel<1><<<gemmGrid, blk256, 0, stream>>>(
        x, wloop1, agg1, b1, h1, NN);

    // ---- layer 2 ----
    for (int r = 0; r < RR; ++r) {
        gemm128_kernel<0><<<gemmGrid, blk256, 0, stream>>>(
            h1, W2 + (size_t)r * HH * HH, nullptr, nullptr, buf, NN);
        edge_scatter_kernel<<<dim3(2048), blk256, 0, stream>>>(
            buf, src, dst, etype, r, agg2);
    }
    // h2 = agg2 + h1 @ wloop2 + b2   (no activation)
    gemm128_kernel<2><<<gemmGrid, blk256, 0, stream>>>(
        h1, wloop2, agg2, b2, h2, NN);

    // ---- head ----
    node_dot_kernel<<<dim3((NN + 255) / 256), blk256, 0, stream>>>(
        h2, wagg, bagg, hv);
    dense1_kernel<<<dim3(100), blk256, 0, stream>>>(hv, wd1, bd1, d1v);
    head_kernel<<<dim3(1), dim3(32), 0, stream>>>(d1v, wd2, bd2, wd3, bd3, out);

    (void)in_sizes; (void)n_in; (void)out_size; (void)ws_size;
}